// ClusterManager_62680752718153
// MI455X (gfx1250) — compile-verified
//
#include <hip/hip_runtime.h>
#include <hip/hip_bf16.h>
#include <math.h>
#include <stdint.h>

#define KDIM 16384   // 64*256 feature dim after reshape
#define C 256        // channels / points
#define B 8          // batch
#define NK 16        // clusters

#define CHUNK 32                 // K elements staged per chunk
#define NCHUNK (KDIM / CHUNK)    // 512
#define LROW 36                  // LDS row stride in floats (144B: 16B aligned, bank-spread)
#define LBUF (16 * LROW)         // floats per 16x32 staged tile (576)

typedef __attribute__((ext_vector_type(2))) float v2f;
typedef __attribute__((ext_vector_type(8))) float v8f;

// raw LDS byte offset of a shared-memory pointer (generic -> AS(3) -> int)
__device__ __forceinline__ uint32_t lds_offset_of(const void* p) {
    return (uint32_t)(uintptr_t)(__attribute__((address_space(3))) const void*)p;
}

// one async 16B global->LDS transfer per lane (32 lanes = 512B per instruction)
__device__ __forceinline__ void async_ld16(uint32_t lds_off, const float* gptr) {
    asm volatile("global_load_async_to_lds_b128 %0, %1, off"
                 :: "v"(lds_off), "v"((uint64_t)(uintptr_t)gptr)
                 : "memory");
}

// ---------------------------------------------------------------------------
// 1) Row norms: norms[b*C+c] = sum_k ff[b,c,k]^2   (2048 rows of 16384)
// ---------------------------------------------------------------------------
__global__ void norms_kernel(const float* __restrict__ ff, float* __restrict__ norms) {
    __shared__ float red[256];
    const int r = blockIdx.x;                 // 0..2047
    const float* row = ff + (size_t)r * KDIM;
    float s = 0.f;
    for (int k = threadIdx.x; k < KDIM; k += 256) {
        float v = row[k];
        s += v * v;
    }
    red[threadIdx.x] = s;
    for (int off = 128; off > 0; off >>= 1) {
        __syncthreads();
        if (threadIdx.x < off) red[threadIdx.x] += red[threadIdx.x + off];
    }
    if (threadIdx.x == 0) norms[r] = red[0];
}

// ---------------------------------------------------------------------------
// 2) Batched Gram + distance: Db[b,i,j] = sqrt(max(n_i + n_j - 2*ff_i.ff_j, 0))
//    One wave per 16x16 tile.  Operands are double-buffered through LDS via
//    global_load_async_to_lds_b128 (ASYNCcnt pipeline), WMMA f32 16x16x4.
//    A 16x4 f32 layout (ISA 7.12.2): lanes 0-15 -> M=lane, K={0,1};
//    lanes 16-31 -> M=lane-16, K={2,3}.  B (4x16) mirrors with N=lane&15.
// ---------------------------------------------------------------------------
__global__ void gram_kernel(const float* __restrict__ ff,
                            const float* __restrict__ norms,
                            float* __restrict__ Db) {
    // per-block LDS: 4 waves x 2 buffers x 2 matrices x LBUF floats
    __shared__ float ldsbuf[4 * 2 * 2 * LBUF];   // 36 KB

    const int b    = blockIdx.z;
    const int ti   = blockIdx.y;
    const int wave = threadIdx.x >> 5;
    const int lane = threadIdx.x & 31;
    const int tj   = blockIdx.x * 4 + wave;
    const int half = lane >> 4;   // 0 or 1
    const int l16  = lane & 15;
    const int i0 = ti * 16, j0 = tj * 16;

    const float* base = ff + (size_t)b * C * KDIM;

    // ---- staging geometry -------------------------------------------------
    // fill: lane covers (row = lane&15, seg = (lane>>4) + 2q), seg = 16B group
    const int frow = l16;
    const int hseg = half;
    const float* gA = base + (size_t)(i0 + frow) * KDIM + hseg * 4;  // + k0 later
    const float* gB = base + (size_t)(j0 + frow) * KDIM + hseg * 4;

    float* wbase = ldsbuf + wave * (4 * LBUF);        // this wave's slice
    float* A_p[2] = { wbase,             wbase + 2 * LBUF };
    float* B_p[2] = { wbase + LBUF,      wbase + 3 * LBUF };

    const uint32_t lane_fill = (uint32_t)(frow * (LROW * 4) + hseg * 16);
    uint32_t lAoff[2] = { lds_offset_of(A_p[0]) + lane_fill,
                          lds_offset_of(A_p[1]) + lane_fill };
    uint32_t lBoff[2] = { lds_offset_of(B_p[0]) + lane_fill,
                          lds_offset_of(B_p[1]) + lane_fill };

    // wmma operand read offset within a staged tile (floats)
    const int rdoff = l16 * LROW + 2 * half;

    // ---- prologue: fill chunk 0 into buffer 0 -----------------------------
    {
        const float* ga = gA;            // k0 = 0
        const float* gb = gB;
#pragma unroll
        for (int q = 0; q < 4; ++q) {
            async_ld16(lAoff[0] + q * 32, ga + q * 8);
            async_ld16(lBoff[0] + q * 32, gb + q * 8);
        }
    }

    v8f acc = {};
    for (int c = 0; c < NCHUNK; ++c) {
        const int p = c & 1;
        if (c + 1 < NCHUNK) {
            const int k1 = (c + 1) * CHUNK;
            const float* ga = gA + k1;
            const float* gb = gB + k1;
            const int pn = p ^ 1;
#pragma unroll
            for (int q = 0; q < 4; ++q) {
                async_ld16(lAoff[pn] + q * 32, ga + q * 8);
                async_ld16(lBoff[pn] + q * 32, gb + q * 8);
            }
            // 8 outstanding (next chunk) -> chunk c has landed (in-order)
            asm volatile("s_wait_asynccnt 0x8" ::: "memory");
        } else {
            asm volatile("s_wait_asynccnt 0x0" ::: "memory");
        }

        const float* Ar = A_p[p] + rdoff;
        const float* Br = B_p[p] + rdoff;
#pragma unroll
        for (int s = 0; s < 8; ++s) {       // 8 x K=4 per 32-K chunk
            v2f a  = *(const v2f*)(Ar + s * 4);
            v2f bb = *(const v2f*)(Br + s * 4);
            acc = __builtin_amdgcn_wmma_f32_16x16x4_f32(
                    /*neg_a=*/false, a, /*neg_b=*/false, bb,
                    /*c_mod=*/(short)0, acc, /*reuse_a=*/false, /*reuse_b=*/false);
        }
    }

    const float* nb = norms + b * C;
    const float  nj = nb[j0 + l16];          // col = lane & 15
    float* Orow = Db + ((size_t)b * C + i0) * C + j0;
#pragma unroll
    for (int r = 0; r < 8; ++r) {
        const int row = r + 8 * half;        // C/D layout: M = r + 8*(lane>=16)
        const float ni = nb[i0 + row];
        const float d2 = ni + nj - 2.0f * acc[r];
        Orow[(size_t)row * C + l16] = sqrtf(fmaxf(d2, 0.0f));
    }
}

// ---------------------------------------------------------------------------
// argmax with first-index tie-break (matches jnp.argmax), 256-thread block
// ---------------------------------------------------------------------------
__device__ __forceinline__ int argmax_red(float* sv, int* si, int tid) {
    for (int off = 128; off > 0; off >>= 1) {
        __syncthreads();
        if (tid < off) {
            float v1 = sv[tid], v2 = sv[tid + off];
            int   i1 = si[tid], i2 = si[tid + off];
            if (v2 > v1 || (v2 == v1 && i2 < i1)) { sv[tid] = v2; si[tid] = i2; }
        }
    }
    __syncthreads();
    int r = si[0];
    __syncthreads();
    return r;
}

// ---------------------------------------------------------------------------
// 3) Farthest-point sampling on each batch's 256x256 distance matrix
// ---------------------------------------------------------------------------
__global__ void fps_batch_kernel(const float* __restrict__ Db, int* __restrict__ sel) {
    __shared__ float sv[256];
    __shared__ int   si[256];
    const int b = blockIdx.x, tid = threadIdx.x;
    const float* D = Db + (size_t)b * C * C;

    float s = 0.f;
    for (int j = 0; j < C; ++j) s += D[(size_t)tid * C + j];
    sv[tid] = s; si[tid] = tid;
    int cur = argmax_red(sv, si, tid);
    if (tid == 0) sel[b * NK + 0] = cur;

    float md = D[(size_t)cur * C + tid];
    for (int t = 1; t < NK; ++t) {
        sv[tid] = md; si[tid] = tid;
        cur = argmax_red(sv, si, tid);
        if (tid == 0) sel[b * NK + t] = cur;
        md = fminf(md, D[(size_t)cur * C + tid]);
        __syncthreads();
    }
}

// ---------------------------------------------------------------------------
// 4) Everything small: FPS on pos, temp assign, segment means, matching,
//    EMA center update, capacity-constrained assignment.  Single block,
//    strictly deterministic (index-ordered sums, no float atomics).
// ---------------------------------------------------------------------------
__global__ void finalize_kernel(const float* __restrict__ pos_emb,
                                const int* __restrict__ sel,
                                int* __restrict__ out) {
    __shared__ float px[C], py[C], pz[C], pn[C];
    __shared__ float sv[C];
    __shared__ int   si[C];
    __shared__ int   cidx[NK];
    __shared__ float ccx[B][NK], ccy[B][NK], ccz[B][NK], ccn[B][NK];
    __shared__ int   ta[B * C];
    __shared__ float ax[NK], ay[NK], az[NK];
    __shared__ float cx[NK], cy[NK], cz[NK];
    __shared__ short ord[C][NK];

    const int tid = threadIdx.x;
    const float x = pos_emb[tid * 3 + 0];
    const float y = pos_emb[tid * 3 + 1];
    const float z = pos_emb[tid * 3 + 2];
    px[tid] = x; py[tid] = y; pz[tid] = z; pn[tid] = x * x + y * y + z * z;
    __syncthreads();

    // --- FPS on pos (row sums need sqrt; later steps use monotone d^2) ---
    float s = 0.f;
    for (int j = 0; j < C; ++j) {
        float d2 = pn[tid] + pn[j] - 2.f * (x * px[j] + y * py[j] + z * pz[j]);
        s += sqrtf(fmaxf(d2, 0.f));
    }
    sv[tid] = s; si[tid] = tid;
    int cur = argmax_red(sv, si, tid);
    if (tid == 0) cidx[0] = cur;
    float md = fmaxf(pn[tid] + pn[cur] - 2.f * (x * px[cur] + y * py[cur] + z * pz[cur]), 0.f);
    for (int t = 1; t < NK; ++t) {
        sv[tid] = md; si[tid] = tid;
        cur = argmax_red(sv, si, tid);
        if (tid == 0) cidx[t] = cur;
        float d2 = fmaxf(pn[tid] + pn[cur] - 2.f * (x * px[cur] + y * py[cur] + z * pz[cur]), 0.f);
        md = fminf(md, d2);
    }
    __syncthreads();

    if (tid < NK) { int c0 = cidx[tid]; cx[tid] = px[c0]; cy[tid] = py[c0]; cz[tid] = pz[c0]; }
    if (tid < B * NK) {
        int bb = tid >> 4, k = tid & 15;
        int c0 = sel[bb * NK + k];
        const float* p = pos_emb + (size_t)(bb * C + c0) * 3;
        float X = p[0], Y = p[1], Z = p[2];
        ccx[bb][k] = X; ccy[bb][k] = Y; ccz[bb][k] = Z; ccn[bb][k] = X * X + Y * Y + Z * Z;
    }
    __syncthreads();

    // --- temp assignment of all B*C points to their batch's centers ---
    for (int p = tid; p < B * C; p += 256) {
        int bb = p >> 8;
        float X = pos_emb[p * 3 + 0], Y = pos_emb[p * 3 + 1], Z = pos_emb[p * 3 + 2];
        float Pn = X * X + Y * Y + Z * Z;
        float best = 3.4e38f; int bi = 0;
        for (int k = 0; k < NK; ++k) {
            float d2 = fmaxf(Pn + ccn[bb][k] - 2.f * (X * ccx[bb][k] + Y * ccy[bb][k] + Z * ccz[bb][k]), 0.f);
            if (d2 < best) { best = d2; bi = k; }   // strict < : first-min tie-break
        }
        ta[p] = bi;
    }
    __syncthreads();

    // --- deterministic segment means: one thread per segment, index order ---
    if (tid < NK) {
        float sx = 0.f, sy = 0.f, sz = 0.f, cntv = 0.f;
        for (int p = 0; p < B * C; ++p) {
            if (ta[p] == tid) {
                sx += pos_emb[p * 3 + 0];
                sy += pos_emb[p * 3 + 1];
                sz += pos_emb[p * 3 + 2];
                cntv += 1.f;
            }
        }
        float denom = fmaxf(cntv, 1.f);
        ax[tid] = cntv > 0.f ? sx / denom : 0.f;
        ay[tid] = cntv > 0.f ? sy / denom : 0.f;
        az[tid] = cntv > 0.f ? sz / denom : 0.f;
    }
    __syncthreads();

    // --- matching + EMA update of centers ---
    if (tid < NK) {
        float X = cx[tid], Y = cy[tid], Z = cz[tid];
        float Cn = X * X + Y * Y + Z * Z;
        float best = 3.4e38f; int bi = 0;
        for (int j = 0; j < NK; ++j) {
            float An = ax[j] * ax[j] + ay[j] * ay[j] + az[j] * az[j];
            float d2 = fmaxf(Cn + An - 2.f * (X * ax[j] + Y * ay[j] + Z * az[j]), 0.f);
            if (d2 < best) { best = d2; bi = j; }
        }
        cx[tid] = 0.8f * X + 0.2f * ax[bi];
        cy[tid] = 0.8f * Y + 0.2f * ay[bi];
        cz[tid] = 0.8f * Z + 0.2f * az[bi];
    }
    __syncthreads();

    // --- per-point stable argsort of distances to updated centers ---
    {
        float dv[NK]; int id[NK];
        for (int k = 0; k < NK; ++k) {
            float Cn = cx[k] * cx[k] + cy[k] * cy[k] + cz[k] * cz[k];
            dv[k] = fmaxf(pn[tid] + Cn - 2.f * (x * cx[k] + y * cy[k] + z * cz[k]), 0.f);
            id[k] = k;
        }
        for (int a = 1; a < NK; ++a) {       // stable insertion sort, ascending
            float v = dv[a]; int ii = id[a]; int bp = a - 1;
            while (bp >= 0 && dv[bp] > v) { dv[bp + 1] = dv[bp]; id[bp + 1] = id[bp]; --bp; }
            dv[bp + 1] = v; id[bp + 1] = ii;
        }
        for (int k = 0; k < NK; ++k) ord[tid][k] = (short)id[k];
    }
    __syncthreads();

    // --- sequential capacity-constrained scan (sizes all == 16) ---
    if (tid == 0) {
        int counts[NK];
        for (int k = 0; k < NK; ++k) counts[k] = 0;
        for (int i = 0; i < C; ++i) {
            int chosen = ord[i][0];
            for (int r = 0; r < NK; ++r) {
                int c0 = ord[i][r];
                if (counts[c0] < 16) { chosen = c0; break; }
            }
            counts[chosen]++;
            out[i] = chosen;
        }
    }
}

// ---------------------------------------------------------------------------
extern "C" void kernel_launch(void* const* d_in, const int* in_sizes, int n_in,
                              void* d_out, int out_size, void* d_ws, size_t ws_size,
                              hipStream_t stream) {
    const float* features = (const float*)d_in[0];   // (8,256,64,256) f32
    const float* pos_emb  = (const float*)d_in[1];   // (8,256,3)      f32

    float* norms = (float*)d_ws;                     // 2048 f32
    float* Db    = norms + B * C;                    // 8*256*256 f32 (2 MB)
    int*   sel   = (int*)(Db + (size_t)B * C * C);   // 8*16 i32
    int*   out   = (int*)d_out;                      // 256 i32 assignments

    norms_kernel<<<B * C, 256, 0, stream>>>(features, norms);
    gram_kernel<<<dim3(4, 16, 8), 128, 0, stream>>>(features, norms, Db);
    fps_batch_kernel<<<B, 256, 0, stream>>>(Db, sel);
    finalize_kernel<<<1, 256, 0, stream>>>(pos_emb, sel, out);
}